// SSTNN_54202487276093
// MI455X (gfx1250) — compile-verified
//
#include <hip/hip_runtime.h>
#include <math.h>

typedef float v2f __attribute__((ext_vector_type(2)));
typedef float v4f __attribute__((ext_vector_type(4)));
typedef float v8f __attribute__((ext_vector_type(8)));

#define I_SIZE   256
#define D_FEAT   128
#define N_DIMS   64
#define N_NEUR   1024
#define T_OUT    256   // only first O_SIZE scan steps are ever observed

#if __has_builtin(__builtin_amdgcn_tanhf)
__device__ __forceinline__ float fast_tanh(float x) { return __builtin_amdgcn_tanhf(x); }
#else
__device__ __forceinline__ float fast_tanh(float x) { return tanhf(x); }
#endif

// ---------------------------------------------------------------------------
// Phase 1 (single WGP, sequential): prologue column-sums + 256-step recurrence
//   u_t = v_t @ W[t]   (128x128 matvec via v_wmma_f32_16x16x4_f32)
//   v_{t+1} = tanh(a[t][t+1] * u_t)
// Stores every u_t (256x128 f32) to workspace for the parallel phase.
// ---------------------------------------------------------------------------
__global__ __launch_bounds__(256) void SSTNN_chain_kernel(
    const float* __restrict__ x,         // (256,128)
    const float* __restrict__ pos_head,  // (1024,64)
    const float* __restrict__ pos_tail,  // (1024,64)
    const float* __restrict__ W,         // (768,128,128)
    const float* __restrict__ a,         // (768,1024)
    float* __restrict__ u_out)           // (256,128) workspace
{
    __shared__ float v_s[D_FEAT];
    __shared__ float u_s[D_FEAT];
    __shared__ float red[N_DIMS];

    const int tid  = threadIdx.x;
    const int lane = tid & 31;
    const int wave = tid >> 5;

    // ---- prologue: v0[j] = (sum_d (sum_k pos_tail[k,d]) * pos_head[0,d]) * sum_i x[i,j]
    float cx = 0.f;
    if (tid < D_FEAT) {
        for (int i = 0; i < I_SIZE; ++i) cx += x[i * D_FEAT + tid];
    }
    if (tid < N_DIMS) {
        float sdp = 0.f;
        for (int k = 0; k < I_SIZE; ++k) sdp += pos_tail[k * N_DIMS + tid];
        red[tid] = sdp * pos_head[tid];   // pos_head row 0
    }
    __syncthreads();
    if (tid == 0) {
        float c0 = 0.f;
        for (int d = 0; d < N_DIMS; ++d) c0 += red[d];
        red[0] = c0;
    }
    __syncthreads();
    if (tid < D_FEAT) v_s[tid] = red[0] * cx;
    __syncthreads();

    // WMMA fragment geometry (wave32):
    //   A 16x4 f32: lanes 0-15 hold K=0,1 in v[0],v[1]; lanes 16-31 hold K=2,3.
    //   B 4x16 f32: lanes 0-15 = N=0..15 with K=0 (v0) / K=1 (v1);
    //               lanes 16-31 = N=0..15 with K=2 (v0) / K=3 (v1).
    //   D 16x16 f32: VGPR0 lanes 0-15 = row M=0  -> acc[0] is the matvec row.
    const int nsub  = (lane < 16) ? lane : (lane - 16);
    const int colB  = wave * 16 + nsub;          // B-matrix column for this lane
    const int koff  = (lane < 16) ? 0 : 2;       // K sub-offset for this half-wave
    const bool row0 = (nsub == 0);               // lane supplies A row M=0

    // Per-lane base pointer into W; all 64 B-fragment loads per step are then
    // compile-time-constant byte offsets (k0*512 / k0*512+512, < 24-bit IOFFSET),
    // so the compiler needs only ONE 64-bit address pair per step.
    const float* __restrict__ Wlane = W + koff * D_FEAT + colB;

    for (int t = 0; t < T_OUT; ++t) {
        const float* __restrict__ Wt = Wlane + (size_t)t * (D_FEAT * D_FEAT);

        v8f acc = {};
#pragma unroll
        for (int k0 = 0; k0 < D_FEAT; k0 += 4) {
            float a0 = v_s[k0 + koff + 0];
            float a1 = v_s[k0 + koff + 1];
            v2f af; af.x = row0 ? a0 : 0.f; af.y = row0 ? a1 : 0.f;
            v2f bf;
            bf.x = Wt[k0 * D_FEAT];            // W[t, k0+koff  , colB]
            bf.y = Wt[k0 * D_FEAT + D_FEAT];   // W[t, k0+koff+1, colB]
            acc = __builtin_amdgcn_wmma_f32_16x16x4_f32(
                      false, af, false, bf, (short)0, acc, false, false);
        }

        // prefetch next step's 64KB weight matrix (256B per thread) AFTER the
        // critical-path loads have issued; overlaps the WMMA chain + barrier.
        if (t + 1 < T_OUT) {
            const float* Wn = W + (size_t)(t + 1) * (D_FEAT * D_FEAT);
#pragma unroll
            for (int p = 0; p < 4; ++p)
                __builtin_prefetch(Wn + tid * 64 + p * 16, 0, 0);
        }

        // row M=0 of D: lanes 0..15, VGPR0 -> u[wave*16 + lane]
        float uval = acc[0];
        if (lane < 16) {
            u_s[wave * 16 + lane] = uval;
            u_out[t * D_FEAT + wave * 16 + lane] = uval;
        }
        __syncthreads();   // all waves done reading v_s and writing u_s

        float s = a[t * N_NEUR + t + 1];            // uniform scalar a[t][t+1]
        if (tid < D_FEAT) v_s[tid] = fast_tanh(s * u_s[tid]);
        __syncthreads();
    }
}

// ---------------------------------------------------------------------------
// Phase 2 (bandwidth-bound): out[t,l,j] = tanh(a[t,l] * u[t,j])
// 128 MiB of coalesced 128-bit nontemporal stores.
// ---------------------------------------------------------------------------
__global__ __launch_bounds__(256) void SSTNN_outer_tanh_kernel(
    const float* __restrict__ a,   // (768,1024), only rows 0..255 used
    const float* __restrict__ u,   // (256,128)
    v4f* __restrict__ out)         // (256,1024,128) as v4f[.. ,32]
{
    const unsigned gid = blockIdx.x * 256u + threadIdx.x;
    const unsigned j4  = gid & 31u;          // float4 index within row (32 per row)
    const unsigned l   = (gid >> 5) & 1023u; // neuron index
    const unsigned t   = gid >> 15;          // step index (wave-uniform -> s_load for a)

    const float s  = a[t * 1024u + l];
    const v4f   uv = ((const v4f*)(u + t * 128u))[j4];

    v4f o;
    o.x = fast_tanh(s * uv.x);
    o.y = fast_tanh(s * uv.y);
    o.z = fast_tanh(s * uv.z);
    o.w = fast_tanh(s * uv.w);
    __builtin_nontemporal_store(o, out + gid);
}

extern "C" void kernel_launch(void* const* d_in, const int* in_sizes, int n_in,
                              void* d_out, int out_size, void* d_ws, size_t ws_size,
                              hipStream_t stream) {
    const float* x  = (const float*)d_in[0];
    const float* ph = (const float*)d_in[1];
    const float* pt = (const float*)d_in[2];
    const float* W  = (const float*)d_in[3];
    const float* a  = (const float*)d_in[4];

    float* u = (float*)d_ws;   // 256*128*4 = 128 KiB of scratch

    SSTNN_chain_kernel<<<1, 256, 0, stream>>>(x, ph, pt, W, a, u);

    // 256*1024*32 v4f elements = 8,388,608 threads
    SSTNN_outer_tanh_kernel<<<32768, 256, 0, stream>>>(a, u, (v4f*)d_out);
}